// SPDNetwork_1_34359738368164
// MI455X (gfx1250) — compile-verified
//
#include <hip/hip_runtime.h>
#include <hip/hip_bf16.h>
#include <math.h>

typedef __attribute__((ext_vector_type(2))) float v2f;
typedef __attribute__((ext_vector_type(8))) float v8f;

#define EPS_REC 1e-4f

// Compiler-only memory ordering barrier. Within one wave32, LDS (DS) ops are
// processed in order by the LDS pipeline, so cross-lane RAW inside a wave only
// needs the compiler not to reorder -- zero hardware cost.
__device__ __forceinline__ void wsync() { __asm__ __volatile__("" ::: "memory"); }

__device__ __forceinline__ v8f wmma4(v2f a, v2f b, v8f c) {
  // (neg_a, A, neg_b, B, c_mod, C, reuse_a, reuse_b)
  return __builtin_amdgcn_wmma_f32_16x16x4_f32(false, a, false, b, (short)0, c,
                                               false, false);
}

// Wave-cooperative GEMM on LDS tiles (row stride 32, 20 rows allocated).
// C[r,c] = sum_{k<K4} opA(r,k) * opB(k,c), result tiles mt_cnt x nt_cnt of
// 16x16, stored masked to rows < 20. K4 must be a multiple of 4, <= 20.
// A fragment layout (f32 16x4): lane<16 holds row=lane, K={k0,k0+1};
// lane>=16 holds row=lane-16, K={k0+2,k0+3}. B mirrors with N on lanes.
// D layout: VGPR r -> row r (lanes 0-15) / row r+8 (lanes 16-31), col = lane&15.
__device__ __forceinline__ void wave_gemm(const float* Ab, bool tA,
                                          const float* Bb, bool tB,
                                          float* Cb,
                                          int mt_cnt, int nt_cnt, int K4,
                                          int lane)
{
  const int lm   = lane & 15;
  const int half = lane >> 4;
  const int kb   = half * 2;
  for (int mt = 0; mt < mt_cnt; ++mt) {
    const int row = mt * 16 + lm;
    for (int nt = 0; nt < nt_cnt; ++nt) {
      const int col = nt * 16 + lm;
      v8f acc = {0.f, 0.f, 0.f, 0.f, 0.f, 0.f, 0.f, 0.f};
      for (int k0 = 0; k0 < K4; k0 += 4) {
        const int ka = k0 + kb;
        v2f av, bv;
        av.x = tA ? Ab[ka * 32 + row]       : Ab[row * 32 + ka];
        av.y = tA ? Ab[(ka + 1) * 32 + row] : Ab[row * 32 + ka + 1];
        bv.x = tB ? Bb[col * 32 + ka]       : Bb[ka * 32 + col];
        bv.y = tB ? Bb[col * 32 + ka + 1]   : Bb[(ka + 1) * 32 + col];
        acc = wmma4(av, bv, acc);
      }
      const int srow = mt * 16 + half * 8;
      #pragma unroll
      for (int r = 0; r < 8; ++r) {
        const int rr = srow + r;
        if (rr < 20) Cb[rr * 32 + col] = acc[r];
      }
    }
  }
  wsync();
}

// Wave-cooperative cyclic Jacobi eigensolver on symmetric n x n matrix in Ab
// (stride 32). Eigenvalues end up on diag(Ab), eigenvectors in columns of Vb.
// Rotations are branchless (safe denominator) so control flow is uniform.
__device__ void wave_jacobi(float* Ab, float* Vb, int n, int sweeps, int lane)
{
  for (int i = lane; i < 640; i += 32) Vb[i] = 0.f;
  wsync();
  if (lane < n) Vb[lane * 32 + lane] = 1.f;
  wsync();
  for (int sw = 0; sw < sweeps; ++sw) {
    for (int p = 0; p < n - 1; ++p) {
      for (int q = p + 1; q < n; ++q) {
        const float apq = Ab[p * 32 + q];
        const float app = Ab[p * 32 + p];
        const float aqq = Ab[q * 32 + q];
        float den = 2.0f * apq;
        den = (fabsf(den) > 1e-30f) ? den : 1e-30f;
        const float tau = (aqq - app) / den;
        const float t =
            ((tau >= 0.f) ? 1.f : -1.f) / (fabsf(tau) + sqrtf(1.f + tau * tau));
        const float c = 1.0f / sqrtf(1.f + t * t);
        const float s = t * c;
        // phase 1: update columns p,q of A and V (A <- A*J, V <- V*J)
        if (lane < n) {
          const int i = lane;
          const float aip = Ab[i * 32 + p], aiq = Ab[i * 32 + q];
          Ab[i * 32 + p] = c * aip - s * aiq;
          Ab[i * 32 + q] = s * aip + c * aiq;
          const float vip = Vb[i * 32 + p], viq = Vb[i * 32 + q];
          Vb[i * 32 + p] = c * vip - s * viq;
          Vb[i * 32 + q] = s * vip + c * viq;
        }
        wsync();
        // phase 2: update rows p,q of A (A <- J^T * A)
        if (lane < n) {
          const int j = lane;
          const float apj = Ab[p * 32 + j], aqj = Ab[q * 32 + j];
          Ab[p * 32 + j] = c * apj - s * aqj;
          Ab[q * 32 + j] = s * apj + c * aqj;
        }
        wsync();
      }
    }
  }
}

__global__ __launch_bounds__(128)
void spdnet_kernel(const float* __restrict__ xg,
                   const float* __restrict__ w1g,
                   const float* __restrict__ w2g,
                   const float* __restrict__ w3g,
                   const float* __restrict__ fcg,
                   float* __restrict__ outg, int Bn)
{
  __shared__ float w1p[640];          // w1 padded into 20x32 (zero pads)
  __shared__ float w2p[640];          // w2 padded into 20x32 (zero pads)
  __shared__ float w3s[64];           // 16x4
  __shared__ float fcs[32];           // 16x2
  __shared__ float bufs[4][3][640];   // per-item working tiles (20x32 each)
  __shared__ float s3[4][80];         // per-item stage-3 scratch

  const int tid  = threadIdx.x;
  const int wid  = tid >> 5;
  const int lane = tid & 31;

  for (int i = tid; i < 640; i += 128) { w1p[i] = 0.f; w2p[i] = 0.f; }
  __syncthreads();
  for (int i = tid; i < 361; i += 128) w1p[(i / 19) * 32 + (i % 19)] = w1g[i];
  for (int i = tid; i < 304; i += 128) w2p[(i / 16) * 32 + (i % 16)] = w2g[i];
  if (tid < 64) w3s[tid] = w3g[tid];
  if (tid < 32) fcs[tid] = fcg[tid];
  __syncthreads();

  const int b = blockIdx.x * 4 + wid;   // wave-uniform
  if (b >= Bn) return;

  float* b0  = bufs[wid][0];
  float* b1  = bufs[wid][1];
  float* b2  = bufs[wid][2];
  float* sc3 = s3[wid];

  // load X[b] (19x19) into zero-padded b0
  for (int i = lane; i < 640; i += 32) b0[i] = 0.f;
  wsync();
  const float* xb = xg + (size_t)b * 361;
  for (int i = lane; i < 361; i += 32) b0[(i / 19) * 32 + (i % 19)] = xb[i];
  wsync();

  // ---- stage 1: Y1 = w1^T X w1, ReEig ----
  wave_gemm(b0, false, w1p, false, b1, 2, 2, 20, lane);  // T1 = X * w1
  wave_gemm(w1p, true, b1, false, b2, 2, 2, 20, lane);   // Y1 = w1^T * T1
  wave_jacobi(b2, b0, 19, 7, lane);                      // b2: diag, b0 = V1
  if (lane < 19) {                                       // W1 = V1 * clamp(S)
    const float scv = fmaxf(b2[lane * 32 + lane], EPS_REC);
    for (int i = 0; i < 19; ++i) b1[i * 32 + lane] = b0[i * 32 + lane] * scv;
  }
  wsync();
  wave_gemm(b1, false, b0, true, b2, 2, 2, 20, lane);    // Z1 = W1 * V1^T

  // ---- stage 2: Y2 = w2^T Z1 w2, ReEig ----
  wave_gemm(b2, false, w2p, false, b1, 2, 1, 20, lane);  // T2 = Z1 * w2
  wave_gemm(w2p, true, b1, false, b0, 1, 1, 20, lane);   // Y2 = w2^T * T2
  wave_jacobi(b0, b2, 16, 7, lane);                      // b0: diag, b2 = V2
  if (lane < 16) {                                       // W2 = V2 * clamp(S)
    const float scv = fmaxf(b0[lane * 32 + lane], EPS_REC);
    for (int i = 0; i < 16; ++i) b1[i * 32 + lane] = b2[i * 32 + lane] * scv;
  }
  wsync();
  wave_gemm(b1, false, b2, true, b0, 1, 1, 16, lane);    // Z2 = W2 * V2^T

  // ---- stage 3: Y3 = w3^T Z2 w3 (4x4) ----
  for (int e = lane; e < 64; e += 32) {                  // T3 = Z2 * w3 (16x4)
    const int r = e >> 2, cc = e & 3;
    float acc = 0.f;
    for (int k = 0; k < 16; ++k) acc += b0[r * 32 + k] * w3s[k * 4 + cc];
    sc3[e] = acc;
  }
  wsync();
  if (lane < 16) {                                       // Y3 = w3^T * T3 (4x4)
    const int r = lane >> 2, cc = lane & 3;
    float acc = 0.f;
    for (int k = 0; k < 16; ++k) acc += w3s[k * 4 + r] * sc3[k * 4 + cc];
    sc3[64 + lane] = acc;
  }
  wsync();

  // ---- redundant per-lane 4x4 eigensolve + matrix log (registers) ----
  float a4[16], v4[16];
  #pragma unroll
  for (int i = 0; i < 16; ++i) {
    a4[i] = sc3[64 + i];
    v4[i] = ((i & 3) == (i >> 2)) ? 1.f : 0.f;
  }
  #pragma unroll
  for (int sw = 0; sw < 6; ++sw) {
    #pragma unroll
    for (int p = 0; p < 3; ++p) {
      #pragma unroll
      for (int q = p + 1; q < 4; ++q) {
        const float apq = a4[p * 4 + q];
        const float app = a4[p * 4 + p];
        const float aqq = a4[q * 4 + q];
        float den = 2.0f * apq;
        den = (fabsf(den) > 1e-30f) ? den : 1e-30f;
        const float tau = (aqq - app) / den;
        const float t = ((tau >= 0.f) ? 1.f : -1.f) /
                        (fabsf(tau) + sqrtf(1.f + tau * tau));
        const float c = 1.0f / sqrtf(1.f + t * t);
        const float s = t * c;
        #pragma unroll
        for (int i = 0; i < 4; ++i) {
          const float aip = a4[i * 4 + p], aiq = a4[i * 4 + q];
          a4[i * 4 + p] = c * aip - s * aiq;
          a4[i * 4 + q] = s * aip + c * aiq;
          const float vip = v4[i * 4 + p], viq = v4[i * 4 + q];
          v4[i * 4 + p] = c * vip - s * viq;
          v4[i * 4 + q] = s * vip + c * viq;
        }
        #pragma unroll
        for (int j = 0; j < 4; ++j) {
          const float apj = a4[p * 4 + j], aqj = a4[q * 4 + j];
          a4[p * 4 + j] = c * apj - s * aqj;
          a4[q * 4 + j] = s * apj + c * aqj;
        }
      }
    }
  }
  float ll[4];
  #pragma unroll
  for (int k = 0; k < 4; ++k) ll[k] = logf(fmaxf(a4[k * 4 + k], 1e-37f));
  float x3[16];
  #pragma unroll
  for (int i = 0; i < 4; ++i) {
    #pragma unroll
    for (int j = 0; j < 4; ++j) {
      float acc = 0.f;
      #pragma unroll
      for (int k = 0; k < 4; ++k) acc += v4[i * 4 + k] * ll[k] * v4[j * 4 + k];
      x3[i * 4 + j] = acc;
    }
  }
  // logits + log_softmax (all lanes redundantly, identical values)
  float l0 = 0.f, l1 = 0.f;
  #pragma unroll
  for (int k = 0; k < 16; ++k) {
    l0 += x3[k] * fcs[k * 2 + 0];
    l1 += x3[k] * fcs[k * 2 + 1];
  }
  const float mx  = fmaxf(l0, l1);
  const float lse = mx + logf(expf(l0 - mx) + expf(l1 - mx));

  if (lane == 0) {
    #pragma unroll
    for (int i = 0; i < 16; ++i) sc3[i] = x3[i];
    outg[(size_t)b * 2 + 0] = l0 - lse;
    outg[(size_t)b * 2 + 1] = l1 - lse;
  }
  wsync();
  if (lane < 16) outg[(size_t)Bn * 2 + (size_t)b * 16 + lane] = sc3[lane];
}

extern "C" void kernel_launch(void* const* d_in, const int* in_sizes, int n_in,
                              void* d_out, int out_size, void* d_ws, size_t ws_size,
                              hipStream_t stream)
{
  const float* xg  = (const float*)d_in[0];
  const float* w1g = (const float*)d_in[1];
  const float* w2g = (const float*)d_in[2];
  const float* w3g = (const float*)d_in[3];
  const float* fcg = (const float*)d_in[4];
  float* outg = (float*)d_out;
  const int Bn = in_sizes[0] / 361;         // 32768
  const int grid = (Bn + 3) / 4;            // 4 items (waves) per block
  spdnet_kernel<<<grid, 128, 0, stream>>>(xg, w1g, w2g, w3g, fcg, outg, Bn);
}